// PointCloudNet_90117003805398
// MI455X (gfx1250) — compile-verified
//
#include <hip/hip_runtime.h>
#include <cstddef>

// PointNet++ segmentation forward for MI455X (gfx1250).
// All channel-MLP GEMMs run on V_WMMA_F32_16X16X4_F32 (fp32 matrix pipe,
// matches reference fp32 numerics). FPS runs fully in LDS with wave32
// shuffle argmax. Activation layout: channel-major [C][B*spatial].
// Workspace requirement: ~241 MB in d_ws.

#define BB 8
#define NN 2048

typedef __attribute__((ext_vector_type(2))) float v2f;
typedef __attribute__((ext_vector_type(8))) float v8f;

// ---------------------------------------------------------------- kernels

// pc [B,N,6] -> pct [6][B*N]
__global__ void k_transpose_pc(const float* __restrict__ pc,
                               float* __restrict__ pct, int BN) {
  int i = blockIdx.x * blockDim.x + threadIdx.x;
  if (i >= BN * 6) return;
  int c = i / BN, bn = i % BN;
  pct[i] = pc[(size_t)bn * 6 + c];
}

// farthest point sampling; one block per batch; xyz strided (6 for raw pc, 3 otherwise)
__global__ void k_fps(const float* __restrict__ xyz, int stride,
                      int* __restrict__ outidx, int N, int npoint) {
  __shared__ float sx[NN], sy[NN], sz[NN], sd[NN];
  __shared__ float wv[32];
  __shared__ int   wi[32];
  __shared__ int   s_last;
  const int b = blockIdx.x, t = threadIdx.x, nt = blockDim.x;
  for (int i = t; i < N; i += nt) {
    const float* p = xyz + ((size_t)b * N + i) * stride;
    sx[i] = p[0]; sy[i] = p[1]; sz[i] = p[2]; sd[i] = 1e10f;
  }
  __syncthreads();
  int last = 0;
  const int nw = nt >> 5;
  for (int it = 0; it < npoint; ++it) {
    if (t == 0) outidx[b * npoint + it] = last;
    float lx = sx[last], ly = sy[last], lz = sz[last];
    float bv = -1.f; int bi = 0;
    for (int i = t; i < N; i += nt) {
      float dx = sx[i] - lx, dy = sy[i] - ly, dz = sz[i] - lz;
      float d  = dx * dx + dy * dy + dz * dz;
      float nd = fminf(sd[i], d);
      sd[i] = nd;
      if (nd > bv) { bv = nd; bi = i; }
    }
    // wave32 argmax (ties -> lowest index, matches jnp.argmax)
    for (int o = 16; o > 0; o >>= 1) {
      float ov = __shfl_down(bv, o);
      int   oi = __shfl_down(bi, o);
      if (ov > bv || (ov == bv && oi < bi)) { bv = ov; bi = oi; }
    }
    if ((t & 31) == 0) { wv[t >> 5] = bv; wi[t >> 5] = bi; }
    __syncthreads();
    if (t == 0) {
      float mv = wv[0]; int mi = wi[0];
      for (int w = 1; w < nw; ++w)
        if (wv[w] > mv || (wv[w] == mv && wi[w] < mi)) { mv = wv[w]; mi = wi[w]; }
      s_last = mi;
    }
    __syncthreads();
    last = s_last;
  }
}

// out[(b*M+m)*3+c] = xyz[(b*Nsrc+idx)*stride+c]
__global__ void k_gather_xyz(const float* __restrict__ xyz, int stride,
                             const int* __restrict__ idx, float* __restrict__ out,
                             int M, int Nsrc, int BM) {
  int i = blockIdx.x * blockDim.x + threadIdx.x;
  if (i >= BM) return;
  int b = i / M;
  int n = idx[i];
  const float* p = xyz + ((size_t)b * Nsrc + n) * stride;
  out[(size_t)i * 3 + 0] = p[0];
  out[(size_t)i * 3 + 1] = p[1];
  out[(size_t)i * 3 + 2] = p[2];
}

// ball query: first S indices (ascending) with d2 < r2; pad with first hit
__global__ void k_ball(const float* __restrict__ xyz, int stride,
                       const float* __restrict__ q, int* __restrict__ out,
                       int BM, int N, int M, int S, float r2) {
  int i = blockIdx.x * blockDim.x + threadIdx.x;
  if (i >= BM) return;
  int b = i / M;
  float qx = q[(size_t)i * 3], qy = q[(size_t)i * 3 + 1], qz = q[(size_t)i * 3 + 2];
  const float* base = xyz + (size_t)b * N * stride;
  int cnt = 0, first = 0;
  for (int n = 0; n < N && cnt < S; ++n) {
    float dx = base[(size_t)n * stride] - qx;
    float dy = base[(size_t)n * stride + 1] - qy;
    float dz = base[(size_t)n * stride + 2] - qz;
    if (dx * dx + dy * dy + dz * dz < r2) {
      if (cnt == 0) first = n;
      out[(size_t)i * S + cnt] = n;
      ++cnt;
    }
  }
  for (; cnt < S; ++cnt) out[(size_t)i * S + cnt] = first;
}

// build grouped input X[3+Cf][B*M*S]: relative xyz + gathered feats
__global__ void k_group(const float* __restrict__ xyz, int stride,
                        const float* __restrict__ nq,
                        const float* __restrict__ feats, int Cf, int fstride,
                        const int* __restrict__ idx, float* __restrict__ X,
                        int Nsrc, int M, int S, int P) {
  int p = blockIdx.x * blockDim.x + threadIdx.x;
  if (p >= P) return;
  int bm = p / S;
  int b  = bm / M;
  int n  = idx[p];
  const float* src = xyz + ((size_t)b * Nsrc + n) * stride;
  const float* c0  = nq + (size_t)bm * 3;
  X[p]                 = src[0] - c0[0];
  X[(size_t)P + p]     = src[1] - c0[1];
  X[(size_t)2 * P + p] = src[2] - c0[2];
  for (int c = 0; c < Cf; ++c)
    X[(size_t)(3 + c) * P + p] = feats[(size_t)c * fstride + (size_t)b * Nsrc + n];
}

// Y[M x P] = W[M x K] * X[K x P] + bias.  One wave = one 16x16 tile.
// fp32 WMMA 16x16x4: A lane l(<16) rows M=tile_m+l, K = k + (hi?2:0) / +1
// B: lanes 0-15 cols N, VGPR0 row K (k+hi*2), VGPR1 row K+1.
// D: VGPR r -> M = tile_m + (hi?8:0) + r, N = tile_n + (lane&15).
// Main K loop is branch-free (no exec-mask churn); only the K%4 tail is
// predicated once.
__global__ void k_gemm_bias(const float* __restrict__ W,
                            const float* __restrict__ bias,
                            const float* __restrict__ X,
                            float* __restrict__ Y,
                            int M, int K, int P) {
  int wave = threadIdx.x >> 5;
  int lane = threadIdx.x & 31;
  long tile_n = ((long)blockIdx.x * (blockDim.x >> 5) + wave) * 16;
  int  tile_m = blockIdx.y * 16;
  if (tile_n >= P) return;           // wave-uniform: EXEC stays all-ones for WMMA
  int  l16 = lane & 15;
  int  hi2 = (lane >> 4) << 1;       // 0 or 2
  int  row = tile_m + l16;
  if (row >= M) row = M - 1;         // clamp; garbage rows never stored
  long col = tile_n + l16;

  const float* pA  = W + (size_t)row * K + hi2;         // A: contiguous pair
  const float* pB0 = X + (size_t)hi2 * P + col;         // B row k+hi2
  const float* pB1 = pB0 + (size_t)P;                   // B row k+hi2+1
  const size_t stepB = (size_t)4 * P;

  v8f acc = {0.f, 0.f, 0.f, 0.f, 0.f, 0.f, 0.f, 0.f};
  int K4 = K & ~3;
  for (int k = 0; k < K4; k += 4) {
    v2f A, Bv;
    A.x  = pA[0];
    A.y  = pA[1];
    Bv.x = pB0[0];
    Bv.y = pB1[0];
    acc = __builtin_amdgcn_wmma_f32_16x16x4_f32(false, A, false, Bv,
                                                (short)0, acc, false, false);
    pA  += 4;
    pB0 += stepB;
    pB1 += stepB;
  }
  if (K4 < K) {                       // single predicated tail step
    int ka = K4 + hi2;
    v2f A, Bv;
    A.x  = (ka < K)     ? pA[0]  : 0.f;
    A.y  = (ka + 1 < K) ? pA[1]  : 0.f;
    Bv.x = (ka < K)     ? pB0[0] : 0.f;
    Bv.y = (ka + 1 < K) ? pB1[0] : 0.f;
    acc = __builtin_amdgcn_wmma_f32_16x16x4_f32(false, A, false, Bv,
                                                (short)0, acc, false, false);
  }
  int mb = tile_m + ((lane >= 16) ? 8 : 0);
  for (int r = 0; r < 8; ++r) {
    int m = mb + r;
    if (m < M) Y[(size_t)m * P + col] = acc[r] + bias[m];
  }
}

// per-channel mean/var over P (batchnorm over batch+spatial)
__global__ void k_bn_stats(const float* __restrict__ Y,
                           float* __restrict__ mean, float* __restrict__ var,
                           size_t P) {
  int c = blockIdx.x, t = threadIdx.x;
  const float* row = Y + (size_t)c * P;
  float s = 0.f, ss = 0.f;
  for (size_t i = t; i < P; i += blockDim.x) { float v = row[i]; s += v; ss += v * v; }
  __shared__ float sh1[256], sh2[256];
  sh1[t] = s; sh2[t] = ss;
  __syncthreads();
  for (int o = 128; o > 0; o >>= 1) {
    if (t < o) { sh1[t] += sh1[t + o]; sh2[t] += sh2[t + o]; }
    __syncthreads();
  }
  if (t == 0) {
    float m = sh1[0] / (float)P;
    mean[c] = m;
    var[c]  = sh2[0] / (float)P - m * m;
  }
}

__global__ void k_bn_act(float* __restrict__ Y,
                         const float* __restrict__ mean, const float* __restrict__ var,
                         const float* __restrict__ g, const float* __restrict__ be,
                         size_t P, size_t tot, int relu) {
  size_t i = (size_t)blockIdx.x * blockDim.x + threadIdx.x;
  if (i >= tot) return;
  int c = (int)(i / P);
  float v = Y[i];
  v = g[c] * (v - mean[c]) * rsqrtf(var[c] + 1e-5f) + be[c];
  if (relu) v = fmaxf(v, 0.f);
  Y[i] = v;
}

// in[c][g*S+s] -> out[c][g] = max over s
__global__ void k_maxpool(const float* __restrict__ Y, float* __restrict__ out,
                          int C, int G, int S) {
  size_t i = (size_t)blockIdx.x * blockDim.x + threadIdx.x;
  if (i >= (size_t)C * G) return;
  const float* p = Y + i * (size_t)S;
  float m = p[0];
  for (int s = 1; s < S; ++s) m = fmaxf(m, p[s]);
  out[i] = m;
}

// 3-NN (smallest squared distances, ties -> lowest index) + interp weights
__global__ void k_knn3(const float* __restrict__ unk, int us,
                       const float* __restrict__ kn,
                       int BM, int M, int Nk,
                       int* __restrict__ idx3, float* __restrict__ w3) {
  int i = blockIdx.x * blockDim.x + threadIdx.x;
  if (i >= BM) return;
  int b = i / M;
  float qx = unk[(size_t)i * us], qy = unk[(size_t)i * us + 1], qz = unk[(size_t)i * us + 2];
  const float* base = kn + (size_t)b * Nk * 3;
  float d0 = 1e30f, d1 = 1e30f, d2 = 1e30f;
  int   i0 = 0, i1 = 0, i2 = 0;
  for (int n = 0; n < Nk; ++n) {
    float dx = base[(size_t)n * 3] - qx;
    float dy = base[(size_t)n * 3 + 1] - qy;
    float dz = base[(size_t)n * 3 + 2] - qz;
    float d = dx * dx + dy * dy + dz * dz;
    if (d < d0)      { d2 = d1; i2 = i1; d1 = d0; i1 = i0; d0 = d; i0 = n; }
    else if (d < d1) { d2 = d1; i2 = i1; d1 = d; i1 = n; }
    else if (d < d2) { d2 = d; i2 = n; }
  }
  float w0 = 1.f / (d0 + 1e-8f), w1 = 1.f / (d1 + 1e-8f), w2 = 1.f / (d2 + 1e-8f);
  float inv = 1.f / (w0 + w1 + w2);
  idx3[(size_t)i * 3] = i0; idx3[(size_t)i * 3 + 1] = i1; idx3[(size_t)i * 3 + 2] = i2;
  w3[(size_t)i * 3] = w0 * inv; w3[(size_t)i * 3 + 1] = w1 * inv; w3[(size_t)i * 3 + 2] = w2 * inv;
}

// dst[c][j] = sum_k feats[c][b*Nk + idx3[j][k]] * w3[j][k]
__global__ void k_interp(const float* __restrict__ f, int fstride,
                         const int* __restrict__ idx3, const float* __restrict__ w3,
                         float* __restrict__ dst, int Pdst, int M, int Nk, int C) {
  size_t i = (size_t)blockIdx.x * blockDim.x + threadIdx.x;
  if (i >= (size_t)C * Pdst) return;
  int c = (int)(i / Pdst);
  int j = (int)(i % Pdst);
  int b = j / M;
  const float* row = f + (size_t)c * fstride + (size_t)b * Nk;
  const int*   ix  = idx3 + (size_t)j * 3;
  const float* w   = w3 + (size_t)j * 3;
  dst[i] = row[ix[0]] * w[0] + row[ix[1]] * w[1] + row[ix[2]] * w[2];
}

// broadcast g_feat[c][b] -> dst[c][b*N+n]
__global__ void k_bcast(const float* __restrict__ g, float* __restrict__ dst,
                        int C, int Bn, int N) {
  size_t tot = (size_t)C * Bn * N;
  size_t i = (size_t)blockIdx.x * blockDim.x + threadIdx.x;
  if (i >= tot) return;
  size_t per = (size_t)Bn * N;
  int c = (int)(i / per);
  size_t r = i % per;
  int b = (int)(r / N);
  dst[i] = g[c * Bn + b];
}

// (B,Cin,Pin) -> (B,Cin/2,2*Pin) row-major reshape, on [C][B*P] layout
__global__ void k_reshape2(const float* __restrict__ in, float* __restrict__ out,
                           int Bn, int Cin, int Pin) {
  size_t tot = (size_t)Bn * Cin * Pin;
  size_t i = (size_t)blockIdx.x * blockDim.x + threadIdx.x;
  if (i >= tot) return;
  int Pout = Pin << 1;
  size_t per = (size_t)Bn * Pout;
  int c2 = (int)(i / per);
  size_t r = i % per;
  int b  = (int)(r / Pout);
  int p2 = (int)(r % Pout);
  int cold = 2 * c2 + (p2 >= Pin ? 1 : 0);
  int pold = (p2 >= Pin) ? (p2 - Pin) : p2;
  out[i] = in[((size_t)cold * Bn + b) * Pin + pold];
}

// Y[3][BP] -> out[bp][3]
__global__ void k_out_t(const float* __restrict__ Y, float* __restrict__ out,
                        size_t BP, int C) {
  size_t i = (size_t)blockIdx.x * blockDim.x + threadIdx.x;
  if (i >= BP * (size_t)C) return;
  int c = (int)(i % C);
  size_t bp = i / C;
  out[i] = Y[(size_t)c * BP + bp];
}

// ---------------------------------------------------------------- host side

struct LayerP { const float *W, *b, *g, *be; };

static const int glob_chs[7]  = {6, 32, 64, 64, 128, 256, 512};
static const int sa_chs[4][4] = {{6, 32, 32, 64}, {67, 64, 64, 128},
                                 {131, 128, 128, 256}, {259, 256, 256, 512}};
static const int fp_chs[4][3] = {{768, 128, 256}, {384, 128, 128},
                                 {192, 128, 128}, {134, 128, 256}};
static const int ups_chs1[4]  = {774, 512, 256, 128};
static const int ups_chs2[3]  = {64, 64, 32};
static const int ups_chs3[2]  = {16, 16};

static void conv_layer(const LayerP& L, int cin, int cout, const float* X, float* Y,
                       long P, float* mean, float* var, int act, int norm,
                       hipStream_t s) {
  dim3 grid((unsigned)((P + 63) / 64), (unsigned)((cout + 15) / 16));
  k_gemm_bias<<<grid, 128, 0, s>>>(L.W, L.b, X, Y, cout, cin, (int)P);
  if (norm) {
    k_bn_stats<<<cout, 256, 0, s>>>(Y, mean, var, (size_t)P);
    size_t tot = (size_t)cout * (size_t)P;
    k_bn_act<<<(unsigned)((tot + 255) / 256), 256, 0, s>>>(Y, mean, var, L.g, L.be,
                                                           (size_t)P, tot, act);
  }
}

static const float* run_mlp(const LayerP* Ls, const int* chs, int nl,
                            const float* X0, float* bufA, float* bufB, long P,
                            float* mean, float* var, hipStream_t s) {
  const float* in = X0;
  float* out = bufA;
  for (int i = 0; i < nl; ++i) {
    conv_layer(Ls[i], chs[i], chs[i + 1], in, out, P, mean, var, 1, 1, s);
    in = out;
    out = (out == bufA) ? bufB : bufA;
  }
  return in;
}

static void sa_module(const float* xyz, int xs, int Nsrc,
                      const float* feats, int Cf,
                      int M, float radius, int S,
                      const LayerP* Ls, const int* chs,
                      float* newxyz, float* outf,
                      int* fpsidx, int* ballidx,
                      float* sA, float* sB, float* mean, float* var,
                      hipStream_t s) {
  k_fps<<<BB, 1024, 0, s>>>(xyz, xs, fpsidx, Nsrc, M);
  int BM = BB * M;
  k_gather_xyz<<<(BM + 255) / 256, 256, 0, s>>>(xyz, xs, fpsidx, newxyz, M, Nsrc, BM);
  k_ball<<<(BM + 255) / 256, 256, 0, s>>>(xyz, xs, newxyz, ballidx, BM, Nsrc, M, S,
                                          radius * radius);
  int P = BM * S;
  k_group<<<(P + 255) / 256, 256, 0, s>>>(xyz, xs, newxyz, feats, Cf, BB * Nsrc,
                                          ballidx, sA, Nsrc, M, S, P);
  const float* y = run_mlp(Ls, chs, 3, sA, sB, sA, (long)P, mean, var, s);
  int Cout = chs[3];
  size_t tot = (size_t)Cout * BM;
  k_maxpool<<<(unsigned)((tot + 255) / 256), 256, 0, s>>>(y, outf, Cout, BM, S);
}

static void fp_module(const float* unk, int us, int Mu,
                      const float* kn, int Nk,
                      const float* feats, int Cf,
                      const float* skip, int Cs,
                      const LayerP* Ls, const int* chs,
                      float* outf, int* knnidx, float* knnw,
                      float* sA, float* sB, float* mean, float* var,
                      hipStream_t s) {
  int BM = BB * Mu;
  k_knn3<<<(BM + 255) / 256, 256, 0, s>>>(unk, us, kn, BM, Mu, Nk, knnidx, knnw);
  size_t it = (size_t)Cf * BM;
  k_interp<<<(unsigned)((it + 255) / 256), 256, 0, s>>>(feats, BB * Nk, knnidx, knnw,
                                                        sA, BM, Mu, Nk, Cf);
  hipMemcpyAsync(sA + (size_t)Cf * BM, skip, (size_t)Cs * BM * sizeof(float),
                 hipMemcpyDeviceToDevice, s);
  const float* y = run_mlp(Ls, chs, 2, sA, sB, sA, (long)BM, mean, var, s);
  hipMemcpyAsync(outf, y, (size_t)chs[2] * BM * sizeof(float),
                 hipMemcpyDeviceToDevice, s);
}

extern "C" void kernel_launch(void* const* d_in, const int* in_sizes, int n_in,
                              void* d_out, int out_size, void* d_ws, size_t ws_size,
                              hipStream_t stream) {
  (void)out_size; (void)ws_size;
  // ---- parameter unpacking: insertion order vs jax-sorted pytree order ----
  // insertion: last tensor = ups_final['b'] (3 elems); sorted: last = pointcloud.
  bool sorted = (in_sizes[n_in - 1] == BB * NN * 6);
  LayerP glob[6], sa[4][3], fp[4][2], ups[6], fin;
  const float* pc = nullptr;
  int cur = 0;
  auto take = [&](LayerP& L) {
    L.W = (const float*)d_in[cur + 0];
    L.b = (const float*)d_in[cur + 1];
    if (sorted) { L.be = (const float*)d_in[cur + 2]; L.g  = (const float*)d_in[cur + 3]; }
    else        { L.g  = (const float*)d_in[cur + 2]; L.be = (const float*)d_in[cur + 3]; }
    cur += 4;
  };
  if (!sorted) {
    pc = (const float*)d_in[0]; cur = 1;
    for (int i = 0; i < 6; ++i) take(glob[i]);
    for (int m = 0; m < 4; ++m) for (int l = 0; l < 3; ++l) take(sa[m][l]);
    for (int m = 0; m < 4; ++m) for (int l = 0; l < 2; ++l) take(fp[m][l]);
    for (int i = 0; i < 6; ++i) take(ups[i]);
    fin.W = (const float*)d_in[cur]; fin.b = (const float*)d_in[cur + 1];
  } else {
    for (int m = 0; m < 4; ++m) for (int l = 0; l < 2; ++l) take(fp[m][l]);
    for (int i = 0; i < 6; ++i) take(glob[i]);
    for (int m = 0; m < 4; ++m) for (int l = 0; l < 3; ++l) take(sa[m][l]);
    for (int i = 0; i < 6; ++i) take(ups[i]);
    fin.W = (const float*)d_in[cur]; fin.b = (const float*)d_in[cur + 1]; cur += 2;
    pc = (const float*)d_in[cur];
  }
  fin.g = nullptr; fin.be = nullptr;

  // ---- workspace carve (floats) ----
  float* base = (float*)d_ws;
  size_t off = 0;
  auto alloc = [&](size_t n) { float* p = base + off; off += n; return p; };
  const size_t BN = (size_t)BB * NN;                  // 16384
  float* sA      = alloc((size_t)32 * 524288);        // ping buffer (max 64 MB)
  float* sB      = alloc((size_t)64 * 524288);        // pong buffer (max 128 MB)
  float* pct     = alloc(6 * BN);
  float* gbuf    = alloc((size_t)512 * BB);
  float* l1x     = alloc(BN * 3);
  float* l2x     = alloc((size_t)BB * 512 * 3);
  float* l3x     = alloc((size_t)BB * 128 * 3);
  float* l4x     = alloc((size_t)BB * 32 * 3);
  float* l1f_sa  = alloc((size_t)64 * BN);
  float* l2f_sa  = alloc((size_t)128 * BB * 512);
  float* l3f_sa  = alloc((size_t)256 * BB * 128);
  float* l4f_sa  = alloc((size_t)512 * BB * 32);
  float* l3f_fp  = alloc((size_t)256 * BB * 128);
  float* l2f_fp  = alloc((size_t)128 * BB * 512);
  float* l1f_fp  = alloc((size_t)128 * BN);
  float* l0f_fp  = alloc((size_t)256 * BN);
  float* meanb   = alloc(512);
  float* varb    = alloc(512);
  float* knnw    = alloc(BN * 3);
  int*   fpsidx  = (int*)alloc(BN);
  int*   ballidx = (int*)alloc(BN * 32);
  int*   knnidx  = (int*)alloc(BN * 3);

  // ---- global feature branch ----
  k_transpose_pc<<<(unsigned)((6 * BN + 255) / 256), 256, 0, stream>>>(pc, pct, (int)BN);
  const float* gy = run_mlp(glob, glob_chs, 6, pct, sA, sB, (long)BN, meanb, varb, stream);
  k_maxpool<<<(unsigned)((512 * BB + 255) / 256), 256, 0, stream>>>(gy, gbuf, 512, BB, NN);

  // ---- set abstraction modules ----
  sa_module(pc, 6, NN, pct + 3 * BN, 3, 2048, 0.1f, 32, sa[0], sa_chs[0],
            l1x, l1f_sa, fpsidx, ballidx, sA, sB, meanb, varb, stream);
  sa_module(l1x, 3, 2048, l1f_sa, 64, 512, 0.2f, 32, sa[1], sa_chs[1],
            l2x, l2f_sa, fpsidx, ballidx, sA, sB, meanb, varb, stream);
  sa_module(l2x, 3, 512, l2f_sa, 128, 128, 0.3f, 32, sa[2], sa_chs[2],
            l3x, l3f_sa, fpsidx, ballidx, sA, sB, meanb, varb, stream);
  sa_module(l3x, 3, 128, l3f_sa, 256, 32, 0.4f, 32, sa[3], sa_chs[3],
            l4x, l4f_sa, fpsidx, ballidx, sA, sB, meanb, varb, stream);

  // ---- feature propagation modules ----
  fp_module(l3x, 3, 128, l4x, 32, l4f_sa, 512, l3f_sa, 256, fp[0], fp_chs[0],
            l3f_fp, knnidx, knnw, sA, sB, meanb, varb, stream);
  fp_module(l2x, 3, 512, l3x, 128, l3f_fp, 256, l2f_sa, 128, fp[1], fp_chs[1],
            l2f_fp, knnidx, knnw, sA, sB, meanb, varb, stream);
  fp_module(l1x, 3, 2048, l2x, 512, l2f_fp, 128, l1f_sa, 64, fp[2], fp_chs[2],
            l1f_fp, knnidx, knnw, sA, sB, meanb, varb, stream);
  fp_module(pc, 6, 2048, l1x, 2048, l1f_fp, 128, pct, 6, fp[3], fp_chs[3],
            l0f_fp, knnidx, knnw, sA, sB, meanb, varb, stream);

  // ---- upsampling head: concat [ip(6), l0f(256), g_feat(512)] = 774 ch ----
  hipMemcpyAsync(sA, pct, 6 * BN * sizeof(float), hipMemcpyDeviceToDevice, stream);
  hipMemcpyAsync(sA + 6 * BN, l0f_fp, (size_t)256 * BN * sizeof(float),
                 hipMemcpyDeviceToDevice, stream);
  k_bcast<<<(unsigned)(((size_t)512 * BN + 255) / 256), 256, 0, stream>>>(
      gbuf, sA + (size_t)262 * BN, 512, BB, NN);

  const float* h1 = run_mlp(ups + 0, ups_chs1, 3, sA, sB, sA, (long)BN, meanb, varb, stream); // ==sB
  k_reshape2<<<(unsigned)(((size_t)128 * BN + 255) / 256), 256, 0, stream>>>(h1, sA, BB, 128, NN);
  const float* h2 = run_mlp(ups + 3, ups_chs2, 2, sA, sB, sA, (long)(BB * 4096),
                            meanb, varb, stream);                                             // ==sA
  k_reshape2<<<(unsigned)(((size_t)32 * BB * 4096 + 255) / 256), 256, 0, stream>>>(h2, sB, BB, 32, 4096);
  const float* h3 = run_mlp(ups + 5, ups_chs3, 1, sB, sA, sB, (long)(BB * 8192),
                            meanb, varb, stream);                                             // ==sA

  // ---- final conv (no norm, no act) + output transpose to [B, 4N, 3] ----
  long Pf = (long)BB * 8192;
  dim3 gf((unsigned)((Pf + 63) / 64), 1);
  k_gemm_bias<<<gf, 128, 0, stream>>>(fin.W, fin.b, h3, sB, 3, 16, (int)Pf);
  size_t tot = (size_t)Pf * 3;
  k_out_t<<<(unsigned)((tot + 255) / 256), 256, 0, stream>>>(sB, (float*)d_out,
                                                             (size_t)Pf, 3);
}